// GNNActor_6425271075459
// MI455X (gfx1250) — compile-verified
//
#include <hip/hip_runtime.h>
#include <hip/hip_bf16.h>
#include <math.h>

typedef float v2f __attribute__((ext_vector_type(2)));
typedef float v8f __attribute__((ext_vector_type(8)));

__device__ __forceinline__ float lrelu(float v) { return v > 0.0f ? v : 0.01f * v; }

// ---------------- degree / normalization ----------------
__global__ void k_degree(const int* __restrict__ dst, float* __restrict__ deg, int E) {
    int e = blockIdx.x * blockDim.x + threadIdx.x;
    if (e < E) atomicAdd(&deg[dst[e]], 1.0f);
}

__global__ void k_dinv(float* __restrict__ deg, int N) {
    int i = blockIdx.x * blockDim.x + threadIdx.x;
    if (i < N) deg[i] = rsqrtf(deg[i] + 1.0f);   // in place: deg -> dinv
}

__global__ void k_normw(const int* __restrict__ src, const int* __restrict__ dst,
                        const float* __restrict__ dinv, float* __restrict__ normw, int E) {
    int e = blockIdx.x * blockDim.x + threadIdx.x;
    if (e < E) normw[e] = dinv[src[e]] * dinv[dst[e]];
}

// ---------------- generic f32 WMMA GEMM: C[N,ncol] = X[N,K] @ W[K,ncol] ----------------
// one 16x16 output tile per wave; blockDim.x = 32 * (ncol/16); grid.x = ceil(N/16)
// act: 0 = raw store (bias may be null), 1 = +bias then leaky-relu(0.01)
__global__ void k_gemm(const float* __restrict__ X, const float* __restrict__ W,
                       const float* __restrict__ bias, float* __restrict__ C,
                       int N, int K, int ncol, int act) {
    const int lane = threadIdx.x & 31;
    const int wave = threadIdx.x >> 5;
    const int row0 = blockIdx.x << 4;
    const int col0 = wave << 4;
    const int m    = lane & 15;            // M for A-frag, N for B/C/D frags
    const int kb   = (lane >> 4) << 1;     // K sub-offset (0 or 2)
    int rowA = row0 + m; if (rowA >= N) rowA = N - 1;   // clamp loads (EXEC stays full)

    v8f acc = {};
    for (int k0 = 0; k0 < K; k0 += 4) {
        const int ka = k0 + kb;
        v2f a, b;
        a.x = X[(size_t)rowA * K + ka];
        a.y = X[(size_t)rowA * K + ka + 1];
        b.x = W[(size_t)ka * ncol + col0 + m];
        b.y = W[(size_t)(ka + 1) * ncol + col0 + m];
        acc = __builtin_amdgcn_wmma_f32_16x16x4_f32(false, a, false, b, (short)0, acc,
                                                    false, false);
    }
    const int nOut  = col0 + m;
    const int rbase = row0 + ((lane >> 4) << 3);
#pragma unroll
    for (int r = 0; r < 8; ++r) {
        const int row = rbase + r;
        if (row < N) {
            float v = acc[r];
            if (act) { v += bias[nOut]; v = lrelu(v); }
            C[(size_t)row * ncol + nOut] = v;
        }
    }
}

// ---------------- MLP layer-1 GEMM with virtual concat A = [out1|out2|out3|state] ----
// C[N,32] = concat[N,256] @ W[256,32]; blockDim.x = 64 (2 waves), grid.x = ceil(N/16)
__global__ void k_mlp1(const float* __restrict__ o1, const float* __restrict__ o2,
                       const float* __restrict__ o3, const float* __restrict__ st,
                       const float* __restrict__ W, const float* __restrict__ bias,
                       float* __restrict__ C, int N) {
    const int lane = threadIdx.x & 31;
    const int wave = threadIdx.x >> 5;
    const int row0 = blockIdx.x << 4;
    const int col0 = wave << 4;
    const int m    = lane & 15;
    const int kb   = (lane >> 4) << 1;
    int rowA = row0 + m; if (rowA >= N) rowA = N - 1;
    const float* const srcs[4] = { o1, o2, o3, st };

    v8f acc = {};
    for (int k0 = 0; k0 < 256; k0 += 4) {
        const int ka = k0 + kb;                  // ka, ka+1 stay inside one 64-wide chunk
        const float* sp = srcs[ka >> 6];
        const int kk = ka & 63;
        v2f a, b;
        a.x = sp[(size_t)rowA * 64 + kk];
        a.y = sp[(size_t)rowA * 64 + kk + 1];
        b.x = W[(size_t)ka * 32 + col0 + m];
        b.y = W[(size_t)(ka + 1) * 32 + col0 + m];
        acc = __builtin_amdgcn_wmma_f32_16x16x4_f32(false, a, false, b, (short)0, acc,
                                                    false, false);
    }
    const int nOut  = col0 + m;
    const int rbase = row0 + ((lane >> 4) << 3);
#pragma unroll
    for (int r = 0; r < 8; ++r) {
        const int row = rbase + r;
        if (row < N)
            C[(size_t)row * 32 + nOut] = lrelu(acc[r] + bias[nOut]);
    }
}

// ---------------- edge scatter: agg[dst] += h[src] * normw  (4 feats/thread) --------
__global__ void k_scatter(const int* __restrict__ src, const int* __restrict__ dst,
                          const float* __restrict__ normw, const float* __restrict__ h,
                          float* __restrict__ agg, int E) {
    long long g = (long long)blockIdx.x * blockDim.x + threadIdx.x;
    if (g >= (long long)E * 16) return;
    const int e  = (int)(g >> 4);
    const int d4 = (int)(g & 15) << 2;
    const int s = src[e], t = dst[e];
    const float w = normw[e];
    const float4 hv = *(const float4*)(h + (size_t)s * 64 + d4);
    float* ap = agg + (size_t)t * 64 + d4;
    atomicAdd(ap + 0, hv.x * w);
    atomicAdd(ap + 1, hv.y * w);
    atomicAdd(ap + 2, hv.z * w);
    atomicAdd(ap + 3, hv.w * w);
}

// ---------------- combine: out = relu(agg + h*dinv^2 + bias) ------------------------
__global__ void k_combine(const float* __restrict__ agg, const float* __restrict__ h,
                          const float* __restrict__ dinv, const float* __restrict__ bias,
                          float* __restrict__ out, int N) {
    long long g = (long long)blockIdx.x * blockDim.x + threadIdx.x;
    if (g < (long long)N * 64) {
        const int i = (int)(g >> 6), d = (int)(g & 63);
        const float di = dinv[i];
        const float v = agg[g] + h[g] * di * di + bias[d];
        out[g] = fmaxf(v, 0.0f);
    }
}

// ---------------- head: conc = softplus(x2 @ lw3 + lb3); accumulate sum -------------
__global__ void k_head(const float* __restrict__ x2, const float* __restrict__ lw3,
                       const float* __restrict__ lb3, float* __restrict__ conc,
                       float* __restrict__ sum, int N) {
    __shared__ float red[256];
    const int i = blockIdx.x * blockDim.x + threadIdx.x;
    float c = 0.0f;
    if (i < N) {
        float s = lb3[0];
        const float* xp = x2 + (size_t)i * 32;
#pragma unroll
        for (int k = 0; k < 32; ++k) s = fmaf(xp[k], lw3[k], s);
        c = (s > 20.0f) ? s : log1pf(expf(s));
        conc[i] = c;
    }
    red[threadIdx.x] = c;
    __syncthreads();
    for (int off = 128; off > 0; off >>= 1) {
        if (threadIdx.x < off) red[threadIdx.x] += red[threadIdx.x + off];
        __syncthreads();
    }
    if (threadIdx.x == 0) atomicAdd(sum, red[0]);
}

// ---------------- finalize: action = conc/(S+eps); regularize = S/N -----------------
__global__ void k_finalize(const float* __restrict__ conc, const float* __restrict__ sum,
                           float* __restrict__ out, int N) {
    const int i = blockIdx.x * blockDim.x + threadIdx.x;
    const float S = *sum;
    if (i < N) out[i] = conc[i] / (S + 1e-20f);
    if (i == 0) out[N] = S / (float)N;
}

extern "C" void kernel_launch(void* const* d_in, const int* in_sizes, int n_in,
                              void* d_out, int out_size, void* d_ws, size_t ws_size,
                              hipStream_t stream) {
    const float* state = (const float*)d_in[0];
    const int*   ei    = (const int*)d_in[1];
    const float* W1 = (const float*)d_in[2];
    const float* b1 = (const float*)d_in[3];
    const float* W2 = (const float*)d_in[4];
    const float* b2 = (const float*)d_in[5];
    const float* W3 = (const float*)d_in[6];
    const float* b3 = (const float*)d_in[7];
    const float* lw1 = (const float*)d_in[8];
    const float* lb1 = (const float*)d_in[9];
    const float* lw2 = (const float*)d_in[10];
    const float* lb2 = (const float*)d_in[11];
    const float* lw3 = (const float*)d_in[12];
    const float* lb3 = (const float*)d_in[13];

    const int N = in_sizes[0] / 64;
    const int E = in_sizes[1] / 2;
    const int* src = ei;
    const int* dst = ei + E;

    // workspace layout (f32)
    float* p = (float*)d_ws;
    float* dinv  = p; p += N;                  // also used as deg before k_dinv
    float* normw = p; p += E;
    float* h     = p; p += (size_t)N * 64;
    float* agg   = p; p += (size_t)N * 64;
    float* out1  = p; p += (size_t)N * 64;
    float* out2  = p; p += (size_t)N * 64;
    float* out3  = p; p += (size_t)N * 64;
    float* x1    = p; p += (size_t)N * 32;
    float* x2    = p; p += (size_t)N * 32;
    float* conc  = p; p += N;
    float* sumb  = p; p += 1;

    const int T = 256;
    const int gE  = (E + T - 1) / T;
    const int gN  = (N + T - 1) / T;
    const int gND = (int)(((long long)N * 64 + T - 1) / T);
    const int gSC = (int)(((long long)E * 16 + T - 1) / T);
    const dim3 gTile((N + 15) / 16);

    // degrees + symmetric norm
    hipMemsetAsync(dinv, 0, (size_t)N * sizeof(float), stream);
    hipMemsetAsync(sumb, 0, sizeof(float), stream);
    k_degree<<<gE, T, 0, stream>>>(dst, dinv, E);
    k_dinv<<<gN, T, 0, stream>>>(dinv, N);
    k_normw<<<gE, T, 0, stream>>>(src, dst, dinv, normw, E);

    // GCN layer 1
    k_gemm<<<gTile, 128, 0, stream>>>(state, W1, nullptr, h, N, 64, 64, 0);
    hipMemsetAsync(agg, 0, (size_t)N * 64 * sizeof(float), stream);
    k_scatter<<<gSC, T, 0, stream>>>(src, dst, normw, h, agg, E);
    k_combine<<<gND, T, 0, stream>>>(agg, h, dinv, b1, out1, N);

    // GCN layer 2
    k_gemm<<<gTile, 128, 0, stream>>>(out1, W2, nullptr, h, N, 64, 64, 0);
    hipMemsetAsync(agg, 0, (size_t)N * 64 * sizeof(float), stream);
    k_scatter<<<gSC, T, 0, stream>>>(src, dst, normw, h, agg, E);
    k_combine<<<gND, T, 0, stream>>>(agg, h, dinv, b2, out2, N);

    // GCN layer 3
    k_gemm<<<gTile, 128, 0, stream>>>(out2, W3, nullptr, h, N, 64, 64, 0);
    hipMemsetAsync(agg, 0, (size_t)N * 64 * sizeof(float), stream);
    k_scatter<<<gSC, T, 0, stream>>>(src, dst, normw, h, agg, E);
    k_combine<<<gND, T, 0, stream>>>(agg, h, dinv, b3, out3, N);

    // MLP head
    k_mlp1<<<gTile, 64, 0, stream>>>(out1, out2, out3, state, lw1, lb1, x1, N);
    k_gemm<<<gTile, 64, 0, stream>>>(x1, lw2, lb2, x2, N, 32, 32, 1);
    k_head<<<gN, T, 0, stream>>>(x2, lw3, lb3, conc, sumb, N);
    k_finalize<<<gN, T, 0, stream>>>(conc, sumb, (float*)d_out, N);
}